// TopKActivation_82566451298891
// MI455X (gfx1250) — compile-verified
//
#include <hip/hip_runtime.h>
#include <stdint.h>

#ifndef __has_builtin
#define __has_builtin(x) 0
#endif

#define TK_COLS     16384
#define TK_THREADS  256

// Pointee type expected by the gfx1250 async global->LDS builtin: int4.
typedef int tk_v4i __attribute__((ext_vector_type(4)));
typedef __attribute__((address_space(1))) tk_v4i tk_v4i_as1;  // global
typedef __attribute__((address_space(3))) tk_v4i tk_v4i_as3;  // LDS

// Monotone float->u32 key: larger float  <=>  larger key (descending select).
__device__ __forceinline__ unsigned tk_key(float f) {
  unsigned u = __float_as_uint(f);
  return (u & 0x80000000u) ? ~u : (u | 0x80000000u);
}

// 16-byte global -> LDS copy via GLOBAL_LOAD_ASYNC_TO_LDS_B128 (ASYNCcnt
// tracked). AS1 pointer = 64-bit VA; AS3 pointer = LDS byte offset, which is
// the low 32 bits of a flat shared pointer (LDS aperture is in the high half).
__device__ __forceinline__ void tk_copy16(const float* g, float* l) {
#if __has_builtin(__builtin_amdgcn_global_load_async_to_lds_b128)
  __builtin_amdgcn_global_load_async_to_lds_b128(
      (tk_v4i_as1*)(uintptr_t)g,
      (tk_v4i_as3*)(unsigned)(uintptr_t)l,
      /*imm offset*/ 0, /*cpol*/ 0);
#else
  *(float4*)l = *(const float4*)g;
#endif
}

__device__ __forceinline__ void tk_wait_async() {
#if __has_builtin(__builtin_amdgcn_global_load_async_to_lds_b128)
#if __has_builtin(__builtin_amdgcn_s_wait_asynccnt)
  __builtin_amdgcn_s_wait_asynccnt(0);
#else
  asm volatile("s_wait_asynccnt 0" ::: "memory");
#endif
#endif
}

// One workgroup per row. Dynamic LDS: row[16384] floats + 256-bin histogram
// + 4 broadcast scalars (66.6 KB; CDNA5 allows up to 320 KB per workgroup).
__global__ __launch_bounds__(TK_THREADS)
void topk_scatter_kernel(const float* __restrict__ x,
                         const int* __restrict__ kptr,
                         float* __restrict__ out) {
  extern __shared__ unsigned char smem[];
  float*    srow = (float*)smem;
  unsigned* hist = (unsigned*)(smem + (size_t)TK_COLS * sizeof(float));
  volatile unsigned* scal = hist + 256;  // [0]=bin [1]=rank-in-bin [2]=bin count

  const int row = blockIdx.x;
  const int tid = threadIdx.x;
  int k = *kptr;
  if (k > TK_COLS) k = TK_COLS;

  const float* grow = x + (size_t)row * TK_COLS;
  float*       orow = out + (size_t)row * TK_COLS;

  if (k <= 0) {  // degenerate: all zeros
    for (int i = tid; i < TK_COLS / 4; i += TK_THREADS)
      ((float4*)orow)[i] = make_float4(0.f, 0.f, 0.f, 0.f);
    return;
  }

  // ---- Stage the row into LDS (HBM read happens exactly once) ----
  for (int i = tid; i < TK_COLS / 4; i += TK_THREADS)
    tk_copy16(grow + 4 * i, srow + 4 * i);
  tk_wait_async();
  __syncthreads();

  // ---- 4x 8-bit MSB-first radix select for the k-th largest key ----
  unsigned prefix = 0u, pmask = 0u;
  int rem = k;      // rank still to satisfy (uniform across block)
  int eqtot = 0;    // element count in the finally selected bin

  for (int pass = 0; pass < 4; ++pass) {
    const int shift = 24 - 8 * pass;

    hist[tid] = 0u;  // TK_THREADS == 256 bins
    __syncthreads();

    // Histogram of keys matching the current prefix (float4 LDS reads).
    for (int i = tid; i < TK_COLS / 4; i += TK_THREADS) {
      float4 v = ((const float4*)srow)[i];
      unsigned u0 = tk_key(v.x), u1 = tk_key(v.y);
      unsigned u2 = tk_key(v.z), u3 = tk_key(v.w);
      if ((u0 & pmask) == prefix) atomicAdd(&hist[(u0 >> shift) & 0xFFu], 1u);
      if ((u1 & pmask) == prefix) atomicAdd(&hist[(u1 >> shift) & 0xFFu], 1u);
      if ((u2 & pmask) == prefix) atomicAdd(&hist[(u2 >> shift) & 0xFFu], 1u);
      if ((u3 & pmask) == prefix) atomicAdd(&hist[(u3 >> shift) & 0xFFu], 1u);
    }
    __syncthreads();

    // Parallel suffix sum over 256 bins (Hillis-Steele, 8 steps):
    // afterwards hist[b] = number of prefix-matching keys with bin >= b.
    #pragma unroll
    for (int off = 1; off < 256; off <<= 1) {
      unsigned v = hist[tid];
      unsigned a = (tid + off < 256) ? hist[tid + off] : 0u;
      __syncthreads();
      hist[tid] = v + a;
      __syncthreads();
    }

    // Exactly one thread's bin brackets rank `rem`:
    //   suffix[b] >= rem  &&  suffix[b+1] < rem
    {
      unsigned suf  = hist[tid];
      unsigned sufn = (tid < 255) ? hist[tid + 1] : 0u;
      if ((unsigned)rem <= suf && (unsigned)rem > sufn) {
        scal[0] = (unsigned)tid;        // selected bin
        scal[1] = (unsigned)rem - sufn; // rank within the bin
        scal[2] = suf - sufn;           // elements in the bin
      }
    }
    __syncthreads();

    prefix |= (scal[0] << shift);
    pmask  |= (0xFFu << shift);
    rem    = (int)scal[1];
    eqtot  = (int)scal[2];
    __syncthreads();  // protect hist before next pass re-zeroes it
  }

  const unsigned thr       = prefix;           // exact key of k-th largest
  const int      needed_eq = rem;              // how many ==thr to keep
  const bool     keep_all  = (needed_eq >= eqtot);  // true unless ties

  // ---- Write pass: HBM write happens exactly once, float4-vectorized ----
  for (int i = tid; i < TK_COLS / 4; i += TK_THREADS) {
    float4 v = ((const float4*)srow)[i];
    unsigned u0 = tk_key(v.x), u1 = tk_key(v.y);
    unsigned u2 = tk_key(v.z), u3 = tk_key(v.w);
    float4 o;
    o.x = (u0 > thr || (keep_all && u0 == thr)) ? v.x : 0.0f;
    o.y = (u1 > thr || (keep_all && u1 == thr)) ? v.y : 0.0f;
    o.z = (u2 > thr || (keep_all && u2 == thr)) ? v.z : 0.0f;
    o.w = (u3 > thr || (keep_all && u3 == thr)) ? v.w : 0.0f;
    ((float4*)orow)[i] = o;
  }

  // ---- Tie-break path (ties at the threshold: keep lowest indices, like
  // jax.lax.top_k). Essentially never taken for continuous random data. ----
  if (!keep_all) {
    __threadfence();
    __syncthreads();
    if (tid == 0) {
      int cnt = 0;
      for (int i = 0; i < TK_COLS && cnt < needed_eq; ++i) {
        if (tk_key(srow[i]) == thr) { orow[i] = srow[i]; ++cnt; }
      }
    }
  }
}

extern "C" void kernel_launch(void* const* d_in, const int* in_sizes, int n_in,
                              void* d_out, int out_size, void* d_ws, size_t ws_size,
                              hipStream_t stream) {
  (void)n_in; (void)out_size; (void)d_ws; (void)ws_size;
  const float* x  = (const float*)d_in[0];
  const int*   kp = (const int*)d_in[1];
  float*       out = (float*)d_out;

  const int rows = in_sizes[0] / TK_COLS;  // 4096 x 16384 fp32
  const size_t smem = (size_t)TK_COLS * sizeof(float)  // row cache (64 KB)
                    + 256 * sizeof(unsigned)           // histogram
                    + 4 * sizeof(unsigned);            // broadcast scalars

  topk_scatter_kernel<<<dim3(rows), dim3(TK_THREADS), smem, stream>>>(x, kp, out);
}